// PSDLoss_12524124635564
// MI455X (gfx1250) — compile-verified
//
#include <hip/hip_runtime.h>

// ---------------------------------------------------------------------------
// PSD spectral loss on gfx1250 (MI455X).
// 2-D DFT as two matmul stages on V_WMMA_F32_16X16X4_F32, with TDM-staged,
// double-buffered LDS (stage A) and async-LDS double buffering (stage B).
//   Stage A: Y(r,i)[256x256] = W(r,i)[256x256] x Gray[256x256]
//   Stage B: Z(r,i)[256x144] = Y x B(r,i)[256x144]   (one-sided, padded 129->144)
// then mag = 20*ln(|Z|^2+eps) -> radial-bin atomics -> normalize -> MSE.
// B operands (Gray, B twiddles) are stored transposed so every WMMA fragment
// load is a contiguous aligned b64 pair.
// ---------------------------------------------------------------------------

typedef __attribute__((ext_vector_type(2))) float v2f;
typedef __attribute__((ext_vector_type(8))) float v8f;
typedef __attribute__((ext_vector_type(4))) unsigned v4u;
typedef __attribute__((ext_vector_type(4))) int v4i;
typedef __attribute__((ext_vector_type(8))) int v8i;

#define WMMA_F32(A, B, C) \
  __builtin_amdgcn_wmma_f32_16x16x4_f32(false, (A), false, (B), (short)0, (C), false, false)

#if __has_builtin(__builtin_amdgcn_tensor_load_to_lds)
#define HAVE_TDM 1
#endif
#if __has_builtin(__builtin_amdgcn_global_load_async_to_lds_b128)
#define HAVE_ASYNC 1
#endif

#define WH       129          // W/2+1
#define WPAD     144          // padded to 9 tiles of 16
#define VLEN     182
#define VPAD     192
#define NIMG     64           // 32 pred + 32 target
#define IMG_ELEM 65536        // 256*256
#define EPSF     1e-8f

// LDS padded row stride (dwords): 32-dword rows + 4 pad -> aligned b64 frag
// loads; 36*m mod 64 distinct over 16 lanes; lane-halves in disjoint banks.
#define LWS 36

// Workspace layout (float element offsets)
#define OFF_WR   0u                    // 65536
#define OFF_WI   65536u                // 65536
#define OFF_BRT  131072u               // 144*256 = 36864 (transposed: [v][y])
#define OFF_BIT  167936u               // 36864
#define OFF_RBIN 204800u               // 36864 (int32 view)
#define OFF_CNT  241664u               // 192
#define OFF_PROF 241856u               // 64*192 = 12288
#define OFF_GRAY 254144u               // 64*65536 (transposed: [n][k])
#define OFF_YR   4448448u              // 64*65536
#define OFF_YI   8642752u              // 64*65536  (end ~51.3 MB)

// ---------------------------------------------------------------------------
#ifdef HAVE_TDM
__device__ __forceinline__ unsigned lds_byte_off(const void* p) {
  return (unsigned)(unsigned long long)(__attribute__((address_space(3))) const char*)p;
}

// 2-D TDM load: 32 contiguous f32 per row, tile_y rows, global stride 256 f32,
// LDS rows padded 32+4 dwords (pad interval code 4 = 32 dwords, amount code 3 = 4).
__device__ __forceinline__ void tdm_load_tile(unsigned lds_byte, const void* gptr,
                                              unsigned tile_y) {
  unsigned long long ga = (unsigned long long)gptr;
  v4u g0;
  g0[0] = 1u;                                   // count=1 (valid user descriptor)
  g0[1] = lds_byte;                             // lds_addr
  g0[2] = (unsigned)ga;                         // global_addr[31:0]
  g0[3] = ((unsigned)(ga >> 32) & 0x01FFFFFFu)  // global_addr[56:32]
          | (2u << 30);                         // type=2 ("image")
  unsigned d0 = (2u << 16)                      // data_size = 4 bytes
              | (1u << 20)                      // pad_enable
              | (4u << 22)                      // pad_interval: 32 dwords
              | (3u << 25);                     // pad_amount: 4 dwords
  v8i g1;
  g1[0] = (int)d0;
  g1[1] = (int)((256u & 0xffffu) << 16);        // tensor_dim0[15:0] = 256
  g1[2] = (int)((tile_y & 0xffffu) << 16);      // td0 hi | tensor_dim1 = tile_y
  g1[3] = (int)(32u << 16);                     // td1 hi | tile_dim0 = 32
  g1[4] = (int)(tile_y & 0xffffu);              // tile_dim1 (tile_dim2 = 0)
  g1[5] = (int)256u;                            // dim0_stride = 256 elements
  g1[6] = 0;
  g1[7] = 0;
  v4i z4 = {0, 0, 0, 0};
  v8i z8 = {0, 0, 0, 0, 0, 0, 0, 0};
  __builtin_amdgcn_tensor_load_to_lds(g0, g1, z4, z4, z8, 0);
}
#endif

#ifdef HAVE_ASYNC
typedef __attribute__((address_space(1))) v4i* gptr_v4i;
typedef __attribute__((address_space(3))) v4i* lptr_v4i;

__device__ __forceinline__ void async_cp16(const float* g, float* l) {
  __builtin_amdgcn_global_load_async_to_lds_b128((gptr_v4i)g, (lptr_v4i)l, 0, 0);
}
#if __has_builtin(__builtin_amdgcn_s_wait_asynccnt)
#define ASYNC_WAIT(n) __builtin_amdgcn_s_wait_asynccnt(n)
#else
#define ASYNC_WAIT(n) asm volatile("s_wait_asynccnt %0" ::"i"(n) : "memory")
#endif
#endif

// ---------------------------------------------------------------------------
// k0: zero counts + profiles (harness poisons ws; re-zero every call)
// ---------------------------------------------------------------------------
__global__ __launch_bounds__(256) void k_zero(float* __restrict__ ws) {
  unsigned id = blockIdx.x * 256u + threadIdx.x;
  const unsigned n = VPAD + NIMG * VPAD;
  if (id < n) ws[OFF_CNT + id] = 0.0f;
}

// ---------------------------------------------------------------------------
// k1: twiddle tables (B transposed), radial-bin LUT, bin counts
// ---------------------------------------------------------------------------
__global__ __launch_bounds__(256) void k_tables(float* __restrict__ ws) {
  unsigned id = blockIdx.x * 256u + threadIdx.x;   // 65536 threads
  int i = id >> 8, j = id & 255;
  float* Wr  = ws + OFF_WR;
  float* Wi  = ws + OFF_WI;
  float* BrT = ws + OFF_BRT;
  float* BiT = ws + OFF_BIT;
  int*   rb  = (int*)(ws + OFF_RBIN);
  float* cnt = ws + OFF_CNT;

  int m = (i * j) & 255;    // phase periodic mod 256 -> exact reduction
  float ang = -6.28318530717958647692f * (float)m * (1.0f / 256.0f);
  float c = cosf(ang), s = sinf(ang);
  Wr[i * 256 + j] = c;
  Wi[i * 256 + j] = s;

  if (j < WPAD) {
    BrT[j * 256 + i] = c;       // [v][y] transposed
    BiT[j * 256 + i] = s;
    if (j < WH) {
      int ii = (i < 128) ? i : i - 256;
      int r = (int)sqrt((double)(ii * ii + j * j));
      rb[i * WPAD + j] = r;
      atomicAdd(&cnt[r], 1.0f);
    } else {
      rb[i * WPAD + j] = 0;
    }
  }
}

// ---------------------------------------------------------------------------
// k2: fused luma + transpose (LDS tiled, both global sides coalesced).
// grayT[img][n][k] = luma(img, row=k, col=n)
// ---------------------------------------------------------------------------
__global__ __launch_bounds__(256) void k_grayT(const float* __restrict__ pred,
                                               const float* __restrict__ target,
                                               float* __restrict__ grayT) {
  __shared__ float tile[32][33];
  const int img = blockIdx.z;
  const int r0 = blockIdx.y * 32;
  const int c0 = blockIdx.x * 32;
  const float* src = (img < 32) ? (pred + (size_t)img * 3u * IMG_ELEM)
                                : (target + (size_t)(img - 32) * 3u * IMG_ELEM);
  const int t = threadIdx.x;
#pragma unroll
  for (int p = 0; p < 4; ++p) {
    int idx = t + p * 256;
    int i = idx >> 5, j = idx & 31;
    int pix = (r0 + i) * 256 + (c0 + j);
    tile[i][j] = 0.299f * src[pix] + 0.587f * src[IMG_ELEM + pix]
               + 0.114f * src[2 * IMG_ELEM + pix];
  }
  __syncthreads();
#pragma unroll
  for (int p = 0; p < 4; ++p) {
    int idx = t + p * 256;
    int i = idx >> 5, j = idx & 31;
    grayT[(size_t)img * IMG_ELEM + (c0 + i) * 256 + (r0 + j)] = tile[j][i];
  }
}

// ---------------------------------------------------------------------------
// k3: stage A — column DFT via f32 WMMA, double-buffered TDM-staged LDS.
// block = 256 threads (8 waves), 64x64 output tile of Yr and Yi.
// ---------------------------------------------------------------------------
__global__ __launch_bounds__(256) void k_dft_cols(const float* __restrict__ grayT,
                                                  const float* __restrict__ Wr,
                                                  const float* __restrict__ Wi,
                                                  float* __restrict__ Yr,
                                                  float* __restrict__ Yi) {
  __shared__ float lWr[2][64 * LWS];
  __shared__ float lWi[2][64 * LWS];
  __shared__ float lXT[2][64 * LWS];   // [n][k] tile of grayT

  const int img = blockIdx.z;
  const int tM = blockIdx.y;       // 0..3
  const int tN = blockIdx.x;       // 0..3
  const int t = threadIdx.x;
  const int wave = t >> 5;
  const int lane = t & 31;
  const int lh = lane >> 4;
  const int lm = lane & 15;
  const int m_off = (wave >> 1) * 16;
  const int n_off = (wave & 1) * 32;
  const float* xT = grayT + (size_t)img * IMG_ELEM;

  v8f ar0 = {}, ar1 = {}, ai0 = {}, ai1 = {};

#ifdef HAVE_TDM
  if (wave == 0) {       // prologue: fill buffer 0 (k0 = 0)
    tdm_load_tile(lds_byte_off(lWr[0]), Wr + (size_t)(tM * 64) * 256, 64);
    tdm_load_tile(lds_byte_off(lWi[0]), Wi + (size_t)(tM * 64) * 256, 64);
    tdm_load_tile(lds_byte_off(lXT[0]), xT + (size_t)(tN * 64) * 256, 64);
  }
#endif

  for (int kb = 0; kb < 8; ++kb) {
    const int cur = kb & 1;
#ifdef HAVE_TDM
    if (wave == 0) {
      if (kb < 7) {      // stream next K-block into the other buffer
        int k0n = (kb + 1) * 32;
        tdm_load_tile(lds_byte_off(lWr[cur ^ 1]), Wr + (size_t)(tM * 64) * 256 + k0n, 64);
        tdm_load_tile(lds_byte_off(lWi[cur ^ 1]), Wi + (size_t)(tM * 64) * 256 + k0n, 64);
        tdm_load_tile(lds_byte_off(lXT[cur ^ 1]), xT + (size_t)(tN * 64) * 256 + k0n, 64);
        __builtin_amdgcn_s_wait_tensorcnt(3);   // oldest batch (cur) done, next in flight
      } else {
        __builtin_amdgcn_s_wait_tensorcnt(0);
      }
    }
    __syncthreads();
#else
    const int k0 = kb * 32;
    __syncthreads();
#pragma unroll
    for (int i = 0; i < 8; ++i) {
      int idx = t + i * 256;
      int r = idx >> 5, c = idx & 31;
      lWr[cur][r * LWS + c] = Wr[(tM * 64 + r) * 256 + k0 + c];
      lWi[cur][r * LWS + c] = Wi[(tM * 64 + r) * 256 + k0 + c];
      lXT[cur][r * LWS + c] = xT[(tN * 64 + r) * 256 + k0 + c];
    }
    __syncthreads();
#endif

    const float* bWr = lWr[cur];
    const float* bWi = lWi[cur];
    const float* bXT = lXT[cur];
#pragma unroll
    for (int kk = 0; kk < 32; kk += 4) {
      // A frag (16x4): lane(lm)=row M, VGPR j -> K = kk + 2*lh + j  (aligned b64)
      int arow = (m_off + lm) * LWS + kk + 2 * lh;
      v2f a_r = *(const v2f*)&bWr[arow];
      v2f a_i = *(const v2f*)&bWi[arow];
      // B frag (4x16): contiguous (K,K+1) pair in transposed X  (aligned b64)
      int brow = (n_off + lm) * LWS + kk + 2 * lh;
      v2f b0 = *(const v2f*)&bXT[brow];
      v2f b1 = *(const v2f*)&bXT[brow + 16 * LWS];

      ar0 = WMMA_F32(a_r, b0, ar0);
      ar1 = WMMA_F32(a_r, b1, ar1);
      ai0 = WMMA_F32(a_i, b0, ai0);
      ai1 = WMMA_F32(a_i, b1, ai1);
    }
    __syncthreads();
  }

  size_t base = (size_t)img * IMG_ELEM;
#pragma unroll
  for (int j = 0; j < 8; ++j) {
    int grow = tM * 64 + m_off + j + 8 * lh;
    int gcol = tN * 64 + n_off + lm;
    Yr[base + grow * 256 + gcol]      = ar0[j];
    Yr[base + grow * 256 + gcol + 16] = ar1[j];
    Yi[base + grow * 256 + gcol]      = ai0[j];
    Yi[base + grow * 256 + gcol + 16] = ai1[j];
  }
}

// ---------------------------------------------------------------------------
// k4: stage B — row DFT (complex, one-sided) + log-power + radial bin.
// block = 128 threads (4 waves), 64 rows x 16 cols of Z.
// Double-buffered async-LDS staging. Zr = Yr*Br - Yi*Bi ; Zi = Yr*Bi + Yi*Br
// ---------------------------------------------------------------------------
__global__ __launch_bounds__(128) void k_dft_rows(const float* __restrict__ Yr,
                                                  const float* __restrict__ Yi,
                                                  const float* __restrict__ BrT,
                                                  const float* __restrict__ BiT,
                                                  const int* __restrict__ rbin,
                                                  float* __restrict__ prof) {
  __shared__ float lYr[2][64 * LWS];
  __shared__ float lYi[2][64 * LWS];
  __shared__ float lBrT[2][16 * LWS];   // [v][k] tile of BrT
  __shared__ float lBiT[2][16 * LWS];

  const int img = blockIdx.z;
  const int u0 = blockIdx.y * 64;
  const int v0 = blockIdx.x * 16;
  const int t = threadIdx.x;
  const int wave = t >> 5;
  const int lane = t & 31;
  const int lh = lane >> 4;
  const int lm = lane & 15;
  const int m_off = wave * 16;
  const size_t base = (size_t)img * IMG_ELEM;

  v8f zr = {}, zi = {};

#ifdef HAVE_ASYNC
  {   // prologue: fill buffer 0 (k0 = 0)
#pragma unroll
    for (int p = 0; p < 4; ++p) {
      int d = (t + p * 128) * 4;
      int r = d >> 5, c = d & 31;
      async_cp16(Yr + base + (u0 + r) * 256 + c, &lYr[0][r * LWS + c]);
      async_cp16(Yi + base + (u0 + r) * 256 + c, &lYi[0][r * LWS + c]);
    }
    int d = t * 4;
    int r = d >> 5, c = d & 31;
    async_cp16(BrT + (v0 + r) * 256 + c, &lBrT[0][r * LWS + c]);
    async_cp16(BiT + (v0 + r) * 256 + c, &lBiT[0][r * LWS + c]);
  }
#endif

  for (int kb = 0; kb < 8; ++kb) {
    const int cur = kb & 1;
#ifdef HAVE_ASYNC
    if (kb < 7) {        // stream next K-block into the other buffer
      int k0n = (kb + 1) * 32;
#pragma unroll
      for (int p = 0; p < 4; ++p) {
        int d = (t + p * 128) * 4;
        int r = d >> 5, c = d & 31;
        async_cp16(Yr + base + (u0 + r) * 256 + k0n + c, &lYr[cur ^ 1][r * LWS + c]);
        async_cp16(Yi + base + (u0 + r) * 256 + k0n + c, &lYi[cur ^ 1][r * LWS + c]);
      }
      int d = t * 4;
      int r = d >> 5, c = d & 31;
      async_cp16(BrT + (v0 + r) * 256 + k0n + c, &lBrT[cur ^ 1][r * LWS + c]);
      async_cp16(BiT + (v0 + r) * 256 + k0n + c, &lBiT[cur ^ 1][r * LWS + c]);
      ASYNC_WAIT(10);    // async loads complete in order: oldest batch (cur) done
    } else {
      ASYNC_WAIT(0);
    }
    __syncthreads();
#else
    const int k0 = kb * 32;
    __syncthreads();
#pragma unroll
    for (int i = 0; i < 16; ++i) {
      int idx = t + i * 128;
      int r = idx >> 5, c = idx & 31;
      lYr[cur][r * LWS + c] = Yr[base + (u0 + r) * 256 + k0 + c];
      lYi[cur][r * LWS + c] = Yi[base + (u0 + r) * 256 + k0 + c];
    }
#pragma unroll
    for (int i = 0; i < 4; ++i) {
      int idx = t + i * 128;
      int r = idx >> 5, c = idx & 31;
      lBrT[cur][r * LWS + c] = BrT[(v0 + r) * 256 + k0 + c];
      lBiT[cur][r * LWS + c] = BiT[(v0 + r) * 256 + k0 + c];
    }
#endif
    __syncthreads();

    const float* bYr = lYr[cur];
    const float* bYi = lYi[cur];
    const float* bBr = lBrT[cur];
    const float* bBi = lBiT[cur];
#pragma unroll
    for (int kk = 0; kk < 32; kk += 4) {
      int arow = (m_off + lm) * LWS + kk + 2 * lh;
      v2f ayr = *(const v2f*)&bYr[arow];
      v2f ayi = *(const v2f*)&bYi[arow];
      int brow = lm * LWS + kk + 2 * lh;
      v2f br = *(const v2f*)&bBr[brow];
      v2f bi = *(const v2f*)&bBi[brow];
      v2f nbi = -bi;   // f32 WMMA has no A/B NEG modifier

      zr = WMMA_F32(ayr, br,  zr);
      zr = WMMA_F32(ayi, nbi, zr);
      zi = WMMA_F32(ayr, bi,  zi);
      zi = WMMA_F32(ayi, br,  zi);
    }
    __syncthreads();
  }

#pragma unroll
  for (int j = 0; j < 8; ++j) {
    int u = u0 + m_off + j + 8 * lh;
    int v = v0 + lm;
    if (v < WH) {
      float mag = 20.0f * logf(zr[j] * zr[j] + zi[j] * zi[j] + EPSF);
      int r = rbin[u * WPAD + v];
      atomicAdd(&prof[img * VPAD + r], mag);
    }
  }
}

// ---------------------------------------------------------------------------
// k5: per-image normalize + MSE (single block).
// ---------------------------------------------------------------------------
__device__ __forceinline__ float block_reduce(float v, int op, float* red) {
  int t = threadIdx.x;
  red[t] = v;
  __syncthreads();
  for (int s = 128; s > 0; s >>= 1) {
    if (t < s) {
      float a = red[t], b = red[t + s];
      red[t] = (op == 0) ? fminf(a, b) : ((op == 1) ? fmaxf(a, b) : (a + b));
    }
    __syncthreads();
  }
  float r = red[0];
  __syncthreads();
  return r;
}

__global__ __launch_bounds__(256) void k_finalize(const float* __restrict__ prof,
                                                  const float* __restrict__ cnt,
                                                  float* __restrict__ out) {
  __shared__ float red[256];
  const int t = threadIdx.x;
  const bool valid = (t < VLEN);
  float accum = 0.0f;

  for (int ip = 0; ip < 32; ++ip) {
    float p = valid ? prof[ip * VPAD + t] / cnt[t] : 0.0f;
    float q = valid ? prof[(ip + 32) * VPAD + t] / cnt[t] : 0.0f;
    float pmin = block_reduce(valid ? p : __builtin_inff(), 0, red);
    float pmax = block_reduce(valid ? p : -__builtin_inff(), 1, red);
    float qmin = block_reduce(valid ? q : __builtin_inff(), 0, red);
    float qmax = block_reduce(valid ? q : -__builtin_inff(), 1, red);
    if (valid) {
      float pn = (p - pmin) / (pmax - pmin);
      float qn = (q - qmin) / (qmax - qmin);
      float d = pn - qn;
      accum += d * d;
    }
  }
  float total = block_reduce(accum, 2, red);
  if (t == 0) out[0] = total / (32.0f * (float)VLEN);
}

// ---------------------------------------------------------------------------
extern "C" void kernel_launch(void* const* d_in, const int* in_sizes, int n_in,
                              void* d_out, int out_size, void* d_ws, size_t ws_size,
                              hipStream_t stream) {
  const float* pred   = (const float*)d_in[0];
  const float* target = (const float*)d_in[1];
  float* ws  = (float*)d_ws;
  float* out = (float*)d_out;

  float* Wr    = ws + OFF_WR;
  float* Wi    = ws + OFF_WI;
  float* BrT   = ws + OFF_BRT;
  float* BiT   = ws + OFF_BIT;
  int*   rbin  = (int*)(ws + OFF_RBIN);
  float* cnt   = ws + OFF_CNT;
  float* prof  = ws + OFF_PROF;
  float* grayT = ws + OFF_GRAY;
  float* Yr    = ws + OFF_YR;
  float* Yi    = ws + OFF_YI;

  k_zero<<<dim3(49), dim3(256), 0, stream>>>(ws);
  k_tables<<<dim3(256), dim3(256), 0, stream>>>(ws);
  k_grayT<<<dim3(8, 8, NIMG), dim3(256), 0, stream>>>(pred, target, grayT);
  k_dft_cols<<<dim3(4, 4, NIMG), dim3(256), 0, stream>>>(grayT, Wr, Wi, Yr, Yi);
  k_dft_rows<<<dim3(9, 4, NIMG), dim3(128), 0, stream>>>(Yr, Yi, BrT, BiT, rbin, prof);
  k_finalize<<<dim3(1), dim3(256), 0, stream>>>(prof, cnt, out);
}